// VectorQuantizerEMA_81140522156340
// MI455X (gfx1250) — compile-verified
//
#include <hip/hip_runtime.h>
#include <math.h>

// MI455X / gfx1250, wave32. f32 WMMA (V_WMMA_F32_16X16X4_F32) for the
// 32768x1024x256 distance GEMM, fused with per-sample argmin.
// B (sample strip) is register-resident; A (code) loads software-pipelined.

typedef __attribute__((ext_vector_type(2))) float v2f;
typedef __attribute__((ext_vector_type(8))) float v8f;

#define N_TOT   32768   // B*H*W
#define D_DIM   256
#define K_CODES 1024
#define HWSZ    1024    // H*W

// ---------------- kernel 0: zero scratch (dw + hist + lossAcc, contiguous)
__global__ void k_init(float* __restrict__ p, int n) {
  int i = blockIdx.x * 256 + threadIdx.x;
  if (i < n) p[i] = 0.f;
}

// ---------------- kernel 1: BCHW -> (N, D) flat, LDS-tiled transpose.
// Per b: transpose (C=256) x (HW=1024). Block: 32(hw) x 32(c) tile, 32x8 threads.
__global__ void __launch_bounds__(256)
k_transpose(const float* __restrict__ in, float* __restrict__ flat) {
  __shared__ float tile[32][33];            // +1 pad: no bank conflicts
  int tx = threadIdx.x;                     // 0..31
  int ty = threadIdx.y;                     // 0..7
  int hwBase = blockIdx.x * 32;             // 32 tiles
  int cBase  = blockIdx.y * 32;             // 8 tiles
  int b      = blockIdx.z;                  // 32
  const float* src = in + ((size_t)b * D_DIM + cBase) * HWSZ + hwBase;
#pragma unroll
  for (int cc = 0; cc < 32; cc += 8)        // coalesced along hw (tx)
    tile[cc + ty][tx] = src[(size_t)(cc + ty) * HWSZ + tx];
  __syncthreads();
  float* dst = flat + ((size_t)b * HWSZ + hwBase) * D_DIM + cBase;
#pragma unroll
  for (int rr = 0; rr < 32; rr += 8)        // coalesced along c (tx)
    dst[(size_t)(rr + ty) * D_DIM + tx] = tile[tx][rr + ty];
}

// ---------------- kernel 2: per-code squared norm
__global__ void k_e2(const float* __restrict__ emb, float* __restrict__ e2) {
  int k = blockIdx.x * 256 + threadIdx.x;
  if (k >= K_CODES) return;
  const float* r = emb + (size_t)k * D_DIM;
  float s = 0.f;
#pragma unroll 8
  for (int d = 0; d < D_DIM; ++d) s = fmaf(r[d], r[d], s);
  e2[k] = s;
}

// ---------------- kernel 3: WMMA distance GEMM + fused argmin
// Wave tile: 16 codes x 16 samples, D=A(16x4 f32)*B(4x16 f32)+C, 64 k-steps.
// C layout: VGPR r, lane L -> code m = r + 8*(L>=16), sample n = L%16.
// argmin(||x||^2+||e||^2-2x.e) over codes == argmin(||e||^2 - 2x.e).
__global__ void __launch_bounds__(256)
k_argmin(const float* __restrict__ flat, const float* __restrict__ emb,
         const float* __restrict__ e2g, int* __restrict__ idxOut) {
  __shared__ float e2s[K_CODES];    // 4KB: code norms, broadcast reads
  int tid = threadIdx.x;
  for (int i = tid; i < K_CODES; i += 256) e2s[i] = e2g[i];
  __syncthreads();

  int lane = tid & 31;
  int wave = tid >> 5;
  int half = lane >> 4;            // 0: K=k0,k0+1  1: K=k0+2,k0+3
  int lq   = lane & 15;            // A: row m=lq ; B: col n=lq
  int sBase = blockIdx.x * 128 + wave * 16;   // 256 blocks * 8 waves * 16 = 32768

  // B strip (this wave's 16 samples, K-slice for this lane) is invariant
  // across all 64 code tiles -> keep resident in 64 v2f = 128 VGPRs.
  const float* bRow = flat + (size_t)(sBase + lq) * D_DIM + (half << 1);
  v2f bs[64];
#pragma unroll
  for (int j = 0; j < 64; ++j) bs[j] = *(const v2f*)(bRow + j * 4);

  float bestVal = 3.4e38f;
  int   bestIdx = 0;

  for (int ct = 0; ct < 64; ++ct) {           // 64 code tiles of 16
    const float* aRow = emb + (size_t)(ct * 16 + lq) * D_DIM + (half << 1);
    if (ct + 1 < 64)                          // global_prefetch_b8 next tile
      __builtin_prefetch(emb + (size_t)((ct + 1) * 16 + lq) * D_DIM, 0, 3);

    v8f acc0 = {};
    v8f acc1 = {};
    // software pipeline: A loads one stage ahead of WMMA consumption
    v2f a0 = *(const v2f*)(aRow);
    v2f a1 = *(const v2f*)(aRow + 4);
#pragma unroll
    for (int j = 0; j < 32; ++j) {            // consumes k = 8j .. 8j+7
      v2f a0n = a0, a1n = a1;
      if (j + 1 < 32) {
        a0n = *(const v2f*)(aRow + (j + 1) * 8);
        a1n = *(const v2f*)(aRow + (j + 1) * 8 + 4);
      }
      acc0 = __builtin_amdgcn_wmma_f32_16x16x4_f32(false, a0, false, bs[2 * j],
                                                   (short)0, acc0, false, false);
      acc1 = __builtin_amdgcn_wmma_f32_16x16x4_f32(false, a1, false, bs[2 * j + 1],
                                                   (short)0, acc1, false, false);
      a0 = a0n; a1 = a1n;
    }
    int mb = ct * 16 + (half << 3);
#pragma unroll
    for (int r = 0; r < 8; ++r) {
      float dot   = acc0[r] + acc1[r];
      float score = e2s[mb + r] - 2.f * dot;  // relative distance
      if (score < bestVal) { bestVal = score; bestIdx = mb + r; }
    }
  }
  // lanes L and L^16 cover disjoint code halves of the same sample column
  float oVal = __shfl_xor(bestVal, 16, 32);
  int   oIdx = __shfl_xor(bestIdx, 16, 32);
  if (oVal < bestVal || (oVal == bestVal && oIdx < bestIdx)) {
    bestVal = oVal; bestIdx = oIdx;
  }
  if (lane < 16) idxOut[sBase + lq] = bestIdx;
}

// ---------------- kernel 4: per-sample outputs (quantized, encodings, dw, hist, loss)
__global__ void __launch_bounds__(256)
k_scatter(const float* __restrict__ flat, const float* __restrict__ emb,
          const int* __restrict__ idxBuf, float* __restrict__ outQ,
          float* __restrict__ outEnc, float* __restrict__ dw,
          float* __restrict__ hist, float* __restrict__ lossAcc) {
  __shared__ float red[256];
  int n = blockIdx.x, c = threadIdx.x;
  int code = idxBuf[n];
  float x = flat[(size_t)n * D_DIM + c];
  float q = emb[(size_t)code * D_DIM + c];
  int b = n >> 10, hw = n & 1023;
  // straight-through: x + sg(q - x) == q numerically; back to BCHW
  outQ[((size_t)(b * D_DIM + c)) * HWSZ + hw] = q;
  atomicAdd(&dw[(size_t)code * D_DIM + c], x);   // dw = enc^T @ flat (scatter-add)
  float d = q - x;
  red[c] = d * d;
  __syncthreads();
  for (int s = 128; s > 0; s >>= 1) {
    if (c < s) red[c] += red[c + s];
    __syncthreads();
  }
  if (c == 0) {
    atomicAdd(lossAcc, red[0]);
    atomicAdd(&hist[code], 1.0f);
  }
  for (int k = c; k < K_CODES; k += 256)
    outEnc[(size_t)n * K_CODES + k] = (k == code) ? 1.f : 0.f;
}

// ---------------- kernel 5: cluster-size normalize + loss + perplexity
__global__ void __launch_bounds__(1024)
k_scalars(const float* __restrict__ emaCS, const float* __restrict__ hist,
          const float* __restrict__ lossAcc, float* __restrict__ outLoss,
          float* __restrict__ outPerp, float* __restrict__ outCS,
          float* __restrict__ csW) {
  __shared__ float red[1024];
  int k = threadIdx.x;
  float raw = 0.99f * emaCS[k] + 0.01f * hist[k];
  red[k] = raw;
  __syncthreads();
  for (int s = 512; s > 0; s >>= 1) { if (k < s) red[k] += red[k + s]; __syncthreads(); }
  float nTot = red[0];
  __syncthreads();
  float p = hist[k] * (1.0f / 32768.f);
  red[k] = -p * logf(p + 1e-10f);
  __syncthreads();
  for (int s = 512; s > 0; s >>= 1) { if (k < s) red[k] += red[k + s]; __syncthreads(); }
  float cs = (raw + 1e-5f) / (nTot + 1024.f * 1e-5f) * nTot;
  outCS[k] = cs;
  csW[k] = cs;
  if (k == 0) {
    outLoss[0] = 0.25f * lossAcc[0] * (1.0f / 8388608.f);
    outPerp[0] = expf(red[0]);
  }
}

// ---------------- kernel 6: EMA weights + normalized embedding
__global__ void k_update(const float* __restrict__ emaW, const float* __restrict__ dw,
                         const float* __restrict__ csW, float* __restrict__ outEma,
                         float* __restrict__ outEmb) {
  int i = blockIdx.x * 256 + threadIdx.x;   // < 262144
  int k = i >> 8;
  float ne = 0.99f * emaW[i] + 0.01f * dw[i];
  outEma[i] = ne;
  float cs = csW[k];
  cs = cs < 1e-5f ? 1e-5f : cs;
  outEmb[i] = ne / cs;
}

extern "C" void kernel_launch(void* const* d_in, const int* in_sizes, int n_in,
                              void* d_out, int out_size, void* d_ws, size_t ws_size,
                              hipStream_t stream) {
  (void)in_sizes; (void)n_in; (void)out_size; (void)ws_size;
  const float* in    = (const float*)d_in[0];  // (B,C,H,W)
  const float* emb   = (const float*)d_in[1];  // (K,D)
  const float* emaW  = (const float*)d_in[2];  // (K,D)
  const float* emaCS = (const float*)d_in[3];  // (K,)
  float* out = (float*)d_out;

  // workspace carve-up (~35 MB)
  // flatW: 8388608 | e2W: 1024 | dwW: 262144 | histW: 1024 | lossW: 1
  // csW: 1024 | idxW: 32768 ints   (dwW..lossW zeroed each call)
  float* flatW = (float*)d_ws;
  float* e2W   = flatW + 8388608;
  float* dwW   = e2W + 1024;
  float* histW = dwW + 262144;
  float* lossW = histW + 1024;
  float* csW   = lossW + 1;
  int*   idxW  = (int*)(csW + 1024);

  // output offsets (tuple concatenated flat, f32)
  float* outLoss = out;                            // 1
  float* outQ    = out + 1;                        // 8388608
  float* outPerp = out + 1 + 8388608;              // 1
  float* outEnc  = out + 2 + 8388608;              // 33554432
  float* outCS   = out + 2 + 8388608 + 33554432;   // 1024
  float* outEma  = outCS + 1024;                   // 262144
  float* outEmb  = outEma + 262144;                // 262144

  k_init<<<(263169 + 255) / 256, 256, 0, stream>>>(dwW, 263169);
  dim3 tgrid(32, 8, 32), tblk(32, 8);
  k_transpose<<<tgrid, tblk, 0, stream>>>(in, flatW);
  k_e2<<<4, 256, 0, stream>>>(emb, e2W);
  k_argmin<<<256, 256, 0, stream>>>(flatW, emb, e2W, idxW);
  k_scatter<<<N_TOT, 256, 0, stream>>>(flatW, emb, idxW, outQ, outEnc, dwW, histW, lossW);
  k_scalars<<<1, 1024, 0, stream>>>(emaCS, histW, lossW, outLoss, outPerp, outCS, csW);
  k_update<<<1024, 256, 0, stream>>>(emaW, dwW, csW, outEma, outEmb);
}